// Net_8048768712829
// MI455X (gfx1250) — compile-verified
//
#include <hip/hip_runtime.h>
#include <stdint.h>

#define NN   100000
#define NE   1600000
#define INF_ 16
#define OF   64
#define PER  12
#define NPH  4
#define FEAT (INF_*PER)      // 192 floats per node
#define NTILES (NN/16)       // 6250 exact

typedef __attribute__((ext_vector_type(2))) float v2f;
typedef __attribute__((ext_vector_type(8))) float v8f;

static __device__ __forceinline__ v8f wmma4(v2f a, v2f b, v8f c) {
  // V_WMMA_F32_16X16X4_F32 : D = A(16x4) * B(4x16) + C(16x16), f32 throughout
  return __builtin_amdgcn_wmma_f32_16x16x4_f32(false, a, false, b, (short)0, c, false, false);
}

// ---------------- degree / norm ----------------
__global__ void k_deg_init(float* __restrict__ deg) {
  int i = blockIdx.x * blockDim.x + threadIdx.x;
  if (i < NN) deg[i] = 1.0f;                 // self-loop contributes 1
}

__global__ void k_deg_edges(const long long* __restrict__ ei, float* __restrict__ deg) {
  int e = blockIdx.x * blockDim.x + threadIdx.x;
  if (e < NE) atomicAdd(&deg[(int)ei[NE + e]], 1.0f);   // dst = edge_index[1]
}

__global__ void k_dinv(float* __restrict__ deg) {
  int i = blockIdx.x * blockDim.x + threadIdx.x;
  if (i < NN) { float d = deg[i]; deg[i] = (d > 0.f) ? rsqrtf(d) : 0.f; }
}

// Y = dinv^2 * X   (self-loop term of A_hat @ X), all 12 periods at once
__global__ void k_init_y(const float* __restrict__ x, const float* __restrict__ dinv,
                         float* __restrict__ Y) {
  long long t = (long long)blockIdx.x * blockDim.x + threadIdx.x;
  if (t < (long long)NN * FEAT) {
    int node = (int)(t / FEAT);
    float di = dinv[node];
    Y[t] = di * di * x[t];
  }
}

// one wave per edge: Y[dst] += norm * X[src]  (192 floats, coalesced f32 atomics)
__global__ void k_scatter(const float* __restrict__ x, const long long* __restrict__ ei,
                          const float* __restrict__ dinv, float* __restrict__ Y) {
  int w    = (blockIdx.x * blockDim.x + threadIdx.x) >> 5;
  int lane = threadIdx.x & 31;
  if (w >= NE) return;
  int s = (int)ei[w];
  int d = (int)ei[NE + w];
  float nrm = dinv[s] * dinv[d];
  const float* xs = x + (long long)s * FEAT;
  float*       yd = Y + (long long)d * FEAT;
#pragma unroll
  for (int c = 0; c < FEAT / 32; ++c) {
    int f = c * 32 + lane;
    atomicAdd(&yd[f], nrm * xs[f]);
  }
}

// ---------------- WMMA node kernel ----------------
// 128 threads = 4 waves, one 16-node tile per wave.
// H0 == 0  =>  Z = sigmoid(convZ @ LZ[:64] + bz'), Ht = tanh(convH @ LH[:64] + bh'),
//              H = (1-Z)*Ht,  Hacc += softmax(attn)[p] * H
// z/h paths fully sequential; a per-period memory clobber stops LICM from
// hoisting ~320 VGPRs of loop-invariant LDS weight fragments (round-2 spills).
__global__ void __launch_bounds__(128, 4)
k_node(const float* __restrict__ Y,
       const float* __restrict__ czw, const float* __restrict__ czb,
       const float* __restrict__ chw, const float* __restrict__ chb,
       const float* __restrict__ lzw, const float* __restrict__ lzb,
       const float* __restrict__ lhw, const float* __restrict__ lhb,
       const float* __restrict__ ow,  const float* __restrict__ ob,
       const float* __restrict__ attn, float* __restrict__ out) {
  __shared__ float s_wz[INF_ * OF], s_wh[INF_ * OF];   // 16x64 conv weights
  __shared__ float s_lz[OF * OF],  s_lh[OF * OF];      // first 64 rows of lin weights
  __shared__ float s_bz[OF], s_bh[OF], s_lzb[OF], s_lhb[OF];
  __shared__ float s_ow[OF * NPH];
  __shared__ float s_ob[NPH];
  __shared__ float s_attn[PER];
  __shared__ float s_tile[4][INF_ * OF];               // per-wave 16x64 scratch

  int tid = threadIdx.x;
  for (int i = tid; i < INF_ * OF; i += 128) { s_wz[i] = czw[i]; s_wh[i] = chw[i]; }
  for (int i = tid; i < OF * OF;  i += 128) { s_lz[i] = lzw[i]; s_lh[i] = lhw[i]; }
  for (int i = tid; i < OF;       i += 128) { s_bz[i] = czb[i]; s_bh[i] = chb[i];
                                              s_lzb[i] = lzb[i]; s_lhb[i] = lhb[i]; }
  for (int i = tid; i < OF * NPH; i += 128) s_ow[i] = ow[i];
  if (tid < NPH) s_ob[tid] = ob[tid];
  if (tid < PER) s_attn[tid] = attn[tid];
  __syncthreads();

  int wave = tid >> 5, lane = tid & 31;
  int tile = blockIdx.x * 4 + wave;
  if (tile >= NTILES) return;

  int node0 = tile * 16;
  int nlo = lane & 15;     // A row / B,C,D column within 16
  int hi  = lane >> 4;     // kh for A/B, mh for C/D
  float* S = s_tile[wave];

  // softmax(attn) scalars (per thread)
  float amax = -1e30f;
  for (int q = 0; q < PER; ++q) amax = fmaxf(amax, s_attn[q]);
  float asum = 0.f;
  for (int q = 0; q < PER; ++q) asum += __expf(s_attn[q] - amax);
  float ainv = 1.0f / asum;

  const v8f z8 = {0.f, 0.f, 0.f, 0.f, 0.f, 0.f, 0.f, 0.f};
  v8f hacc[4] = {z8, z8, z8, z8};

  const float* Yb = Y + (long long)(node0 + nlo) * FEAT;  // row base for this lane

  for (int p = 0; p < PER; ++p) {
    // Stop LICM from hoisting LDS weight-fragment loads out of the period loop
    // (they total ~320 VGPRs and spill otherwise). Re-reading LDS is cheap.
    asm volatile("" ::: "memory");

    float prob = __expf(s_attn[p] - amax) * ainv;

    // ================= z path =================
    v8f acc[4] = {z8, z8, z8, z8};
    for (int kk = 0; kk < 4; ++kk) {                 // convZ: (16x16)@(16x64)
      int k0 = 4 * kk + 2 * hi;
      v2f a; a.x = Yb[k0 * PER + p]; a.y = Yb[(k0 + 1) * PER + p];
#pragma unroll
      for (int nt = 0; nt < 4; ++nt) {
        v2f b; b.x = s_wz[k0 * OF + nt * 16 + nlo]; b.y = s_wz[(k0 + 1) * OF + nt * 16 + nlo];
        acc[nt] = wmma4(a, b, acc[nt]);
      }
    }
#pragma unroll
    for (int nt = 0; nt < 4; ++nt)
#pragma unroll
      for (int v = 0; v < 8; ++v)
        S[(v + 8 * hi) * OF + nt * 16 + nlo] = acc[nt][v] + s_bz[nt * 16 + nlo];

    v8f zz[4] = {z8, z8, z8, z8};
    for (int kk = 0; kk < 16; ++kk) {                // Zpre: (16x64)@(64x64)
      int k0 = 4 * kk + 2 * hi;
      v2f a; a.x = S[nlo * OF + k0]; a.y = S[nlo * OF + k0 + 1];
#pragma unroll
      for (int nt = 0; nt < 4; ++nt) {
        v2f b; b.x = s_lz[k0 * OF + nt * 16 + nlo]; b.y = s_lz[(k0 + 1) * OF + nt * 16 + nlo];
        zz[nt] = wmma4(a, b, zz[nt]);
      }
    }
#pragma unroll
    for (int nt = 0; nt < 4; ++nt)
#pragma unroll
      for (int v = 0; v < 8; ++v) {
        float zv = zz[nt][v] + s_lzb[nt * 16 + nlo];
        zz[nt][v] = 1.0f / (1.0f + __expf(-zv));     // Z
      }

    // ================= h path (reuse acc + LDS tile) =================
#pragma unroll
    for (int nt = 0; nt < 4; ++nt) acc[nt] = z8;
    for (int kk = 0; kk < 4; ++kk) {                 // convH
      int k0 = 4 * kk + 2 * hi;
      v2f a; a.x = Yb[k0 * PER + p]; a.y = Yb[(k0 + 1) * PER + p];
#pragma unroll
      for (int nt = 0; nt < 4; ++nt) {
        v2f b; b.x = s_wh[k0 * OF + nt * 16 + nlo]; b.y = s_wh[(k0 + 1) * OF + nt * 16 + nlo];
        acc[nt] = wmma4(a, b, acc[nt]);
      }
    }
#pragma unroll
    for (int nt = 0; nt < 4; ++nt)
#pragma unroll
      for (int v = 0; v < 8; ++v)
        S[(v + 8 * hi) * OF + nt * 16 + nlo] = acc[nt][v] + s_bh[nt * 16 + nlo];

#pragma unroll
    for (int nt = 0; nt < 4; ++nt) acc[nt] = z8;
    for (int kk = 0; kk < 16; ++kk) {                // Hpre
      int k0 = 4 * kk + 2 * hi;
      v2f a; a.x = S[nlo * OF + k0]; a.y = S[nlo * OF + k0 + 1];
#pragma unroll
      for (int nt = 0; nt < 4; ++nt) {
        v2f b; b.x = s_lh[k0 * OF + nt * 16 + nlo]; b.y = s_lh[(k0 + 1) * OF + nt * 16 + nlo];
        acc[nt] = wmma4(a, b, acc[nt]);
      }
    }
#pragma unroll
    for (int nt = 0; nt < 4; ++nt)
#pragma unroll
      for (int v = 0; v < 8; ++v) {
        float t = tanhf(acc[nt][v] + s_lhb[nt * 16 + nlo]);
        hacc[nt][v] += prob * (1.0f - zz[nt][v]) * t;  // H0==0 => H=(1-Z)*Ht
      }
  }

  // ---- output head: relu -> (64x4) matvec -> softmax over 4 ----
#pragma unroll
  for (int nt = 0; nt < 4; ++nt)
#pragma unroll
    for (int v = 0; v < 8; ++v)
      S[(v + 8 * hi) * OF + nt * 16 + nlo] = fmaxf(hacc[nt][v], 0.0f);

  if (lane < 16) {
    int node = node0 + lane;
    float lg[NPH];
#pragma unroll
    for (int c = 0; c < NPH; ++c) lg[c] = s_ob[c];
    for (int f = 0; f < OF; ++f) {
      float hv = S[lane * OF + f];
#pragma unroll
      for (int c = 0; c < NPH; ++c) lg[c] += hv * s_ow[f * NPH + c];
    }
    float mx = fmaxf(fmaxf(lg[0], lg[1]), fmaxf(lg[2], lg[3]));
    float sm = 0.f;
#pragma unroll
    for (int c = 0; c < NPH; ++c) { lg[c] = __expf(lg[c] - mx); sm += lg[c]; }
    float inv = 1.0f / sm;
#pragma unroll
    for (int c = 0; c < NPH; ++c) out[(long long)node * NPH + c] = lg[c] * inv;
  }
}

// ---------------- launch ----------------
extern "C" void kernel_launch(void* const* d_in, const int* in_sizes, int n_in,
                              void* d_out, int out_size, void* d_ws, size_t ws_size,
                              hipStream_t stream) {
  (void)in_sizes; (void)n_in; (void)out_size; (void)ws_size;
  const float*     x    = (const float*)d_in[0];
  const long long* ei   = (const long long*)d_in[1];
  const float*     attn = (const float*)d_in[2];
  const float*     czw  = (const float*)d_in[3];
  const float*     czb  = (const float*)d_in[4];
  // d_in[5], d_in[6] (conv_r_*) are dead: H0 stays zero in the reference.
  const float*     chw  = (const float*)d_in[7];
  const float*     chb  = (const float*)d_in[8];
  const float*     lzw  = (const float*)d_in[9];
  const float*     lzbv = (const float*)d_in[10];
  // d_in[11], d_in[12] (lin_r_*) dead for the same reason.
  const float*     lhw  = (const float*)d_in[13];
  const float*     lhbv = (const float*)d_in[14];
  const float*     ow   = (const float*)d_in[15];
  const float*     obv  = (const float*)d_in[16];
  float*           out  = (float*)d_out;

  float* dinv = (float*)d_ws;                         // NN floats (deg -> dinv in place)
  float* Y    = dinv + ((NN + 255) & ~255);           // NN*192 floats

  k_deg_init <<<(NN + 255) / 256, 256, 0, stream>>>(dinv);
  k_deg_edges<<<(NE + 255) / 256, 256, 0, stream>>>(ei, dinv);
  k_dinv     <<<(NN + 255) / 256, 256, 0, stream>>>(dinv);
  long long tot = (long long)NN * FEAT;
  k_init_y   <<<(int)((tot + 255) / 256), 256, 0, stream>>>(x, dinv, Y);
  k_scatter  <<<(NE + 7) / 8, 256, 0, stream>>>(x, ei, dinv, Y);
  k_node     <<<(NTILES + 3) / 4, 128, 0, stream>>>(Y, czw, czb, chw, chb,
                                                    lzw, lzbv, lhw, lhbv,
                                                    ow, obv, attn, out);
}